// ATCA_TCA_Loss_46316927320609
// MI455X (gfx1250) — compile-verified
//
#include <hip/hip_runtime.h>
#include <stdint.h>

#define TT 500
#define NN 128
#define CC 10
#define KMAX 46   // T // (C+1) + 1

typedef unsigned int v4u __attribute__((ext_vector_type(4)));
typedef int v8i __attribute__((ext_vector_type(8)));
typedef int v4i __attribute__((ext_vector_type(4)));

__global__ __launch_bounds__(NN)
void atca_tca_kernel(const float* __restrict__ vmem,
                     const int* __restrict__ labels,
                     float* __restrict__ out,
                     float* __restrict__ partial) {
  extern __shared__ float smem[];          // [TT*NN] trace | [KMAX*NN] cluster maxima | [NN] reduce
  float* recs = smem + TT * NN;
  float* red  = recs + KMAX * NN;

  const int b = blockIdx.x;
  const int n = threadIdx.x;               // lane <-> neuron

  // -------- TDM: stage vmem[b,:,:] (500x128 f32 = 256000 B, contiguous) into LDS --------
  // One tensor_load_to_lds per wave; wave w moves bytes [w*64000, (w+1)*64000).
  {
    const unsigned wave =
        ((unsigned)__builtin_amdgcn_readfirstlane((int)threadIdx.x)) >> 5;
    const uint64_t gaddr = (uint64_t)(uintptr_t)(vmem + (size_t)b * (TT * NN))
                         + (uint64_t)wave * 64000u;
    const uint32_t laddr = (uint32_t)(uintptr_t)smem + wave * 64000u;

    v4u g0;
    g0[0] = 1u;                                       // count=1 (valid user D#)
    g0[1] = laddr;                                    // lds_addr (bytes)
    g0[2] = (uint32_t)(gaddr & 0xFFFFFFFFu);          // global_addr[31:0]
    g0[3] = (uint32_t)((gaddr >> 32) & 0x01FFFFFFu)   // global_addr[56:32]
          | 0x80000000u;                              // type = 2 ("image")

    v8i g1;
    g1[0] = 0x00020000;                  // data_size=2 -> 4 B; no multicast/pad/iterate
    g1[1] = (int)(16000u << 16);         // tensor_dim0[15:0]=16000 in bits[63:48]
    g1[2] = 0x00010000;                  // tensor_dim1 = 1
    g1[3] = (int)(16000u << 16);         // tile_dim0 = 16000 in bits[127:112]
    g1[4] = 0x00000001;                  // tile_dim1 = 1, tile_dim2 = 0
    g1[5] = 16000;                       // tensor_dim0_stride[31:0]
    g1[6] = 0;
    g1[7] = 0;

    v4i gz4 = {0, 0, 0, 0};              // groups 2/3: >2D dims unused
    v8i gz8 = {0, 0, 0, 0, 0, 0, 0, 0};  // trailing group (6-arg toolchain variant)

    __builtin_amdgcn_tensor_load_to_lds(g0, g1, gz4, gz4, gz8, 0);
    __builtin_amdgcn_s_wait_tensorcnt(0);
  }
  __syncthreads();

  const float* tr = smem + n;            // this neuron's trace, stride NN (bank-conflict-free)

  // -------- pass 1: vmax / first argmax, cluster count, per-cluster span maxima --------
  float vmax = -3.0e38f;
  int   tm = 0;
  int   nc = 0;
  int   lastSpike = -0x40000000;
  float pending = -3.0e38f;              // max of v over (lastSpike, t]
  float clusterMax = 0.0f;
  for (int t = 0; t < TT; ++t) {
    const float x = tr[t * NN];
    if (x > vmax) { vmax = x; tm = t; }
    pending = fmaxf(pending, x);
    if (x >= 0.0f) {                     // spike
      if (t - lastSpike > CC) {          // starts a new cluster
        if (nc > 0) recs[(nc - 1) * NN + n] = clusterMax;
        ++nc;
        clusterMax = x;                  // span restarts at this spike
      } else {
        clusterMax = fmaxf(clusterMax, pending);  // span extends through (lastSpike, t]
      }
      pending = -3.0e38f;
      lastSpike = t;
    }
  }
  if (nc > 0) recs[(nc - 1) * NN + n] = clusterMax;

  // -------- pass 2 (C-delayed scan): masked maxima for the excess branch --------
  // At delayed index tp = t-C, lastSpike == latest spike <= tp+C, so
  // mask[tp] (= spike in [tp-C, tp+C]) <=> lastSpike >= tp-C.
  const int wlo = tm - CC / 2, whi = tm + CC / 2;
  float m0 = -1.0e30f, mr = -1.0e30f;
  int full0 = 1, full2 = 1;
  lastSpike = -0x40000000;
  for (int t = 0; t < TT + CC; ++t) {
    if (t < TT) {
      if (tr[t * NN] >= 0.0f) lastSpike = t;
    }
    const int tp = t - CC;
    if (tp >= 0) {
      const int mask  = (lastSpike >= tp - CC);
      const int mask2 = mask | ((tp >= wlo) & (tp <= whi));
      full0 &= mask;
      full2 &= mask2;
      if (!mask2) {                      // !mask2 implies !mask
        const float vp = tr[tp * NN];
        m0 = fmaxf(m0, vp);
        mr = fmaxf(mr, vp);
      } else if (!mask) {
        m0 = fmaxf(m0, tr[tp * NN]);
      }
    }
  }

  // -------- branch combine --------
  const int label = labels[b * NN + n];
  float contrib = 0.0f;
  if (label > nc) {
    const float dE = (float)(label - nc);            // >= 1
    const float mrest = full2 ? vmax : mr;
    const float mean_excess = (m0 + (dE - 1.0f) * mrest) / dE;
    contrib = full0 ? vmax : -mean_excess;
  } else if (label < nc) {
    const int m = nc - label;                        // sum of m smallest cluster maxima
    float s = 0.0f;
    for (int i = 0; i < m; ++i) {
      float best = 3.0e38f; int bi = 0;
      for (int k = 0; k < nc; ++k) {
        const float r = recs[k * NN + n];
        if (r < best) { best = r; bi = k; }
      }
      s += best;
      recs[bi * NN + n] = 3.0e38f;                   // mark taken
    }
    contrib = s / (float)m;
  }

  out[1 + b * NN + n] = (float)nc;                   // spike_output[b, n]

  // -------- deterministic block reduction of the loss contribution --------
  red[n] = contrib;
  __syncthreads();
  #pragma unroll
  for (int s = NN / 2; s > 0; s >>= 1) {
    if (n < s) red[n] += red[n + s];
    __syncthreads();
  }
  if (n == 0) partial[b] = red[0];
}

__global__ __launch_bounds__(256)
void atca_sum_kernel(const float* __restrict__ partial, float* __restrict__ out, int nb) {
  __shared__ float red[256];
  const int i = threadIdx.x;
  red[i] = (i < nb) ? partial[i] : 0.0f;
  __syncthreads();
  #pragma unroll
  for (int s = 128; s > 0; s >>= 1) {
    if (i < s) red[i] += red[i + s];
    __syncthreads();
  }
  if (i == 0) out[0] = red[0];
}

extern "C" void kernel_launch(void* const* d_in, const int* in_sizes, int n_in,
                              void* d_out, int out_size, void* d_ws, size_t ws_size,
                              hipStream_t stream) {
  const float* vmem   = (const float*)d_in[0];   // [B, T, N] f32
  // d_in[1] (vlastmem) and d_in[3] (ratio) are unused by the reference forward.
  const int*   labels = (const int*)d_in[2];     // [B, N] i32
  float* out     = (float*)d_out;                // [1 + B*N] : loss, then cluster counts
  float* partial = (float*)d_ws;                 // [B] block partial sums
  const int B = in_sizes[0] / (TT * NN);         // 256

  const size_t shmem = (size_t)(TT * NN + KMAX * NN + NN) * sizeof(float); // 280,064 B < 320 KB
  hipLaunchKernelGGL(atca_tca_kernel, dim3(B), dim3(NN), shmem, stream,
                     vmem, labels, out, partial);
  hipLaunchKernelGGL(atca_sum_kernel, dim3(1), dim3(256), 0, stream, partial, out, B);
}